// SS_Block_79147657331153
// MI455X (gfx1250) — compile-verified
//
#include <hip/hip_runtime.h>
#include <stdint.h>

// ---------------------------------------------------------------------------
// 2-D Haar transform, (16,64,512,512) f32 -> 4x (16,64,256,256) f32.
// Bandwidth-bound (~2.15 GB traffic, ~1 GFLOP -> ~92 us at 23.3 TB/s).
// Strategy:
//   * one wave per "row-pair" (input rows 2h,2h+1 are contiguous: 4 KB chunk)
//   * CDNA5 async data mover: global_load_async_to_lds_b128 (NT hint),
//     double-buffered per wave via s_wait_asynccnt (copy next / compute cur)
//   * compute from LDS, emit 16B non-temporal stores (TH_STORE_NT)
// ---------------------------------------------------------------------------

#define WPB 4                       // waves per block (blockDim = 128)
#define ROWPAIR_FLOATS 1024         // two contiguous input rows of 512 f32
#define ROWPAIR_BYTES  4096
#define BUF_FLOATS (2 * ROWPAIR_FLOATS)  // double buffer per wave (8 KB)

// Native clang vector type: __builtin_nontemporal_store accepts this
// (HIP's float4 is a struct and is rejected).
typedef float v4f __attribute__((ext_vector_type(4)));

// Issue 8 async b128 loads copying one contiguous 4 KB row-pair into LDS.
// INST_OFFSET is added to BOTH the LDS and global address (ISA 08 §4.4),
// so one (lds, global) register pair serves all 8 chunks.
__device__ __forceinline__ void async_copy_rowpair(const float* __restrict__ src,
                                                   uint32_t lds_byte, int lane) {
    uint64_t ga = (uint64_t)(uintptr_t)src + (uint32_t)(lane * 16);
    uint32_t la = lds_byte + (uint32_t)(lane * 16);
    asm volatile(
        "global_load_async_to_lds_b128 %0, %1, off th:TH_LOAD_NT\n\t"
        "global_load_async_to_lds_b128 %0, %1, off offset:512 th:TH_LOAD_NT\n\t"
        "global_load_async_to_lds_b128 %0, %1, off offset:1024 th:TH_LOAD_NT\n\t"
        "global_load_async_to_lds_b128 %0, %1, off offset:1536 th:TH_LOAD_NT\n\t"
        "global_load_async_to_lds_b128 %0, %1, off offset:2048 th:TH_LOAD_NT\n\t"
        "global_load_async_to_lds_b128 %0, %1, off offset:2560 th:TH_LOAD_NT\n\t"
        "global_load_async_to_lds_b128 %0, %1, off offset:3072 th:TH_LOAD_NT\n\t"
        "global_load_async_to_lds_b128 %0, %1, off offset:3584 th:TH_LOAD_NT"
        :
        : "v"(la), "v"(ga)
        : "memory");
}

// Async loads complete in order: with 16 outstanding (8 old + 8 new),
// waiting to <=8 guarantees the old buffer is fully in LDS.
__device__ __forceinline__ void wait_async_le8() {
    asm volatile("s_wait_asynccnt 0x8" ::: "memory");
}
__device__ __forceinline__ void wait_async_le0() {
    asm volatile("s_wait_asynccnt 0x0" ::: "memory");
}

// Lane computes output cols [8l, 8l+8) of one output row from LDS rows r0/r1.
__device__ __forceinline__ void haar_row(const float* __restrict__ r0,
                                         const float* __restrict__ r1,
                                         int lane,
                                         float* __restrict__ oA,
                                         float* __restrict__ oH,
                                         float* __restrict__ oV,
                                         float* __restrict__ oD) {
    const int c0 = lane * 8;
#pragma unroll
    for (int h = 0; h < 2; ++h) {
        const int cb = c0 + h * 4;
        v4f A, H, V, D;
#pragma unroll
        for (int e = 0; e < 4; ++e) {
            const int c = cb + e;
            const float x00 = r0[2 * c];
            const float x01 = r0[2 * c + 1];
            const float x10 = r1[2 * c];
            const float x11 = r1[2 * c + 1];
            const float s = x00 + x01;
            const float t = x10 + x11;
            const float u = x00 - x01;
            const float w = x10 - x11;
            A[e] = 0.5f * (s + t);
            H[e] = 0.5f * (s - t);
            V[e] = 0.5f * (u + w);
            D[e] = 0.5f * (u - w);
        }
        __builtin_nontemporal_store(A, (v4f*)(oA + cb));
        __builtin_nontemporal_store(H, (v4f*)(oH + cb));
        __builtin_nontemporal_store(V, (v4f*)(oV + cb));
        __builtin_nontemporal_store(D, (v4f*)(oD + cb));
    }
}

__global__ __launch_bounds__(WPB * 32) void haar2d_async_kernel(
    const float* __restrict__ x, float* __restrict__ out,
    int rp_total, long long q) {
    __shared__ float smem[WPB * BUF_FLOATS];  // 32 KB: 4 waves x 8 KB dbl-buf

    const int lane = threadIdx.x & 31;
    const int wave = threadIdx.x >> 5;
    const int wstride = gridDim.x * WPB;

    float* smem_wave = &smem[wave * BUF_FLOATS];
    // Generic pointers into LDS carry the LDS byte offset in addr[31:0]
    // async ops take that offset.
    const uint32_t lds_base = (uint32_t)(uintptr_t)smem_wave;

    int rp = blockIdx.x * WPB + wave;   // wave-uniform row-pair index
    if (rp >= rp_total) return;         // uniform exit, EXEC stays all-ones

    // Prologue: fill buffer 0.
    async_copy_rowpair(x + (size_t)rp * ROWPAIR_FLOATS, lds_base, lane);

    int buf = 0;
    for (;;) {
        const int rpn = rp + wstride;
        const bool have_next = rpn < rp_total;
        if (have_next) {
            // Prefetch next row-pair into the other buffer while we compute.
            async_copy_rowpair(x + (size_t)rpn * ROWPAIR_FLOATS,
                               lds_base + (uint32_t)((buf ^ 1) * ROWPAIR_BYTES),
                               lane);
            wait_async_le8();   // oldest 8 (current buffer) are complete
        } else {
            wait_async_le0();   // drain
        }

        const float* r0 = smem_wave + buf * ROWPAIR_FLOATS;
        const float* r1 = r0 + 512;
        const size_t ob = (size_t)rp * 256;
        haar_row(r0, r1, lane,
                 out + ob,
                 out + (size_t)q + ob,
                 out + 2 * (size_t)q + ob,
                 out + 3 * (size_t)q + ob);

        if (!have_next) break;
        rp = rpn;
        buf ^= 1;
    }
}

extern "C" void kernel_launch(void* const* d_in, const int* in_sizes, int n_in,
                              void* d_out, int out_size, void* d_ws, size_t ws_size,
                              hipStream_t stream) {
    (void)in_sizes; (void)n_in; (void)out_size; (void)d_ws; (void)ws_size;
    const float* x = (const float*)d_in[0];
    float* out = (float*)d_out;

    const int rp_total = 16 * 64 * 256;          // 262144 row-pairs
    const long long q = 16LL * 64 * 256 * 256;   // quadrant stride (elements)

    // 8192 blocks x 4 waves = 32768 waves -> 8 pipelined iterations per wave.
    dim3 grid(8192), block(WPB * 32);
    hipLaunchKernelGGL(haar2d_async_kernel, grid, block, 0, stream,
                       x, out, rp_total, q);
}